// MecanumSystemModel_23819888624084
// MI455X (gfx1250) — compile-verified
//
#include <hip/hip_runtime.h>
#include <stdint.h>

typedef float v4f __attribute__((ext_vector_type(4)));

// ---------------- model constants ----------------
// Computed in double then cast to float, matching the reference's
// float64 -> float32 constant pipeline. pinv(l2w) is analytic because the
// columns of l2w are exactly orthogonal: pinv = diag(1/4,1/4,1/(4h^2)) @ l2w^T.
constexpr double dG    = 13.7;
constexpr double dR    = 3.0 * 0.0254;
constexpr double dMASS = 12.0;
constexpr double dMOI  = dMASS * (12.0 * 0.0254) * (12.0 * 0.0254) / 6.0;
constexpr double dH    = 0.048 + 0.128;

constexpr float G_F   = (float)dG;
constexpr float INVR  = (float)(1.0 / dR);
constexpr float H_F   = (float)dH;
constexpr float K0    = (float)(1.0 / (4.0 * dR * dMASS));
constexpr float K2    = (float)(1.0 / (4.0 * dH * dR * dMOI));
constexpr float KVISC = (float)(dG * 0.000304);
constexpr float KMOT  = 0.193f;
constexpr float KSS   = 0.00317f;
constexpr float EPS_F = 0.01f;

__device__ __forceinline__ float wheel_torque(float md, float wv) {
  float ss = wv * rsqrtf(fmaf(wv, wv, EPS_F));   // softsign
  return (fmaf(md, KMOT, -wv * KVISC) - ss * KSS) * G_F;
}

// out = [vx, vy, omega, accel_x, accel_y, accel_omega]
__device__ __forceinline__ void mecanum_row(float theta, float vx, float vy, float om,
                                            float u0, float u1, float u2, float o[6]) {
  float s = __sinf(theta);
  float c = __cosf(theta);
  float lx = fmaf(c, vx,  s * vy);           // rotate by -theta
  float ly = fmaf(c, vy, -s * vx);
  float hw = H_F * om;
  float wv0 = (lx - ly - hw) * INVR;
  float wv1 = (lx + ly + hw) * INVR;
  float wv2 = (lx + ly - hw) * INVR;
  float wv3 = (lx - ly + hw) * INVR;
  float md0 = u0 - u1 - u2;
  float md1 = u0 + u1 + u2;
  float md2 = u0 + u1 - u2;
  float md3 = u0 - u1 + u2;
  float t0 = wheel_torque(md0, wv0);
  float t1 = wheel_torque(md1, wv1);
  float t2 = wheel_torque(md2, wv2);
  float t3 = wheel_torque(md3, wv3);
  float lax = K0 * ((t0 + t1) + (t2 + t3));
  float lay = K0 * ((t1 + t2) - (t0 + t3));
  float laz = K2 * ((t1 + t3) - (t0 + t2));
  float aax = fmaf(c, lax, -s * lay);        // rotate by +theta
  float aay = fmaf(s, lax,  c * lay);
  o[0] = vx;  o[1] = vy;  o[2] = om;
  o[3] = aax; o[4] = aay; o[5] = laz;
}

// ---------------- async-staged main kernel ----------------
// One quad (4 rows) per lane; one 32-quad tile per wave.
// Staging: per-wave GLOBAL_LOAD_ASYNC_TO_LDS_B128 — each instruction moves
// 32 lanes x 16B = 512B *contiguous* global (4 cache lines/instr, perfect
// coalescing), scattered into a padded per-quad LDS layout.
// State quad = 96B padded to 112B (stride 28 dwords, gcd(28,64)=4 -> 2-way
// bank conflicts on ds_load_b128). Ctrl quad = 48B (stride 12 dwords, 2-way).
constexpr int WAVES_PER_BLOCK = 8;
constexpr int S_STRIDE_B = 112;                       // per-quad state bytes in LDS
constexpr int C_STRIDE_B = 48;                        // per-quad ctrl bytes in LDS
constexpr int WAVE_LDS_B = 32 * S_STRIDE_B + 32 * C_STRIDE_B;   // 5120 B

__global__ __launch_bounds__(WAVES_PER_BLOCK * 32)
void mecanum_async(const float* __restrict__ state, const float* __restrict__ ctrl,
                   float* __restrict__ out, int nfullwave) {
  __shared__ __align__(16) float lds[WAVES_PER_BLOCK * WAVE_LDS_B / 4];

  const int lane  = threadIdx.x & 31;
  const int wib   = threadIdx.x >> 5;                 // wave in block
  const int gwave = blockIdx.x * WAVES_PER_BLOCK + wib;
  if (gwave >= nfullwave) return;                     // wave-uniform guard

  // Absolute LDS byte address of this wave's staging region.
  // (Generic LDS pointer: addr[31:0] is the LDS byte address.)
  const uint32_t ldsBase  = (uint32_t)(uintptr_t)(&lds[0]);
  const uint32_t sRegion  = ldsBase + (uint32_t)wib * WAVE_LDS_B;
  const uint32_t cRegion  = sRegion + 32 * S_STRIDE_B;

  const size_t waveQuad = (size_t)gwave * 32;
  const unsigned long long sBase =
      (unsigned long long)(uintptr_t)state + waveQuad * 96ull;
  const unsigned long long cBase =
      (unsigned long long)(uintptr_t)ctrl + waveQuad * 48ull;

  // ---- stage state: 192 float4 / wave, 6 async b128 per lane group ----
#pragma unroll
  for (int k = 0; k < 6; ++k) {
    uint32_t g    = (uint32_t)(k * 32 + lane);        // local float4 idx [0,192)
    uint32_t q    = g / 6u, j = g % 6u;               // dest quad, slot
    uint32_t dst  = sRegion + q * S_STRIDE_B + j * 16u;
    uint32_t goff = g * 16u;
    asm volatile("global_load_async_to_lds_b128 %0, %1, %2"
                 :: "v"(dst), "v"(goff), "s"(sBase) : "memory");
  }
  // ---- stage control: 96 float4 / wave, 3 async b128 per lane group ----
#pragma unroll
  for (int k = 0; k < 3; ++k) {
    uint32_t g    = (uint32_t)(k * 32 + lane);        // [0,96)
    uint32_t q    = g / 3u, j = g % 3u;
    uint32_t dst  = cRegion + q * C_STRIDE_B + j * 16u;
    uint32_t goff = g * 16u;
    asm volatile("global_load_async_to_lds_b128 %0, %1, %2"
                 :: "v"(dst), "v"(goff), "s"(cBase) : "memory");
  }
  // Wave-local staging: only ASYNCcnt drain needed, no barrier.
  asm volatile("s_wait_asynccnt 0x0" ::: "memory");

  // ---- read own quad back from LDS (ds_load_b128, 2-way conflicts) ----
  const int sF = wib * (WAVE_LDS_B / 4) + lane * (S_STRIDE_B / 4);
  const int cF = wib * (WAVE_LDS_B / 4) + 32 * (S_STRIDE_B / 4) + lane * (C_STRIDE_B / 4);
  const v4f* sQ = (const v4f*)&lds[sF];
  const v4f* cQ = (const v4f*)&lds[cF];

  float st[24], ct[12];
#pragma unroll
  for (int k = 0; k < 6; ++k) {
    v4f v = sQ[k];
#pragma unroll
    for (int j = 0; j < 4; ++j) st[k * 4 + j] = v[j];
  }
#pragma unroll
  for (int k = 0; k < 3; ++k) {
    v4f v = cQ[k];
#pragma unroll
    for (int j = 0; j < 4; ++j) ct[k * 4 + j] = v[j];
  }

  float ot[24];
#pragma unroll
  for (int r = 0; r < 4; ++r) {
    mecanum_row(st[6 * r + 2], st[6 * r + 3], st[6 * r + 4], st[6 * r + 5],
                ct[3 * r + 0], ct[3 * r + 1], ct[3 * r + 2], &ot[6 * r]);
  }

  v4f* op = (v4f*)out + (waveQuad + (size_t)lane) * 6;
#pragma unroll
  for (int k = 0; k < 6; ++k) {
    v4f v;
#pragma unroll
    for (int j = 0; j < 4; ++j) v[j] = ot[k * 4 + j];
    __builtin_nontemporal_store(v, op + k);            // NT b128 stores
  }
}

// ---------------- fallback: direct-load quad kernel (covers quad tail) ----------------
__global__ __launch_bounds__(256)
void mecanum_quad(const float* __restrict__ state, const float* __restrict__ ctrl,
                  float* __restrict__ out, int startQuad, int nquad) {
  int i = startQuad + blockIdx.x * blockDim.x + threadIdx.x;
  if (i >= nquad) return;
  const v4f* sp = (const v4f*)state + (size_t)i * 6;
  const v4f* cp = (const v4f*)ctrl  + (size_t)i * 3;
  v4f sv[6];
#pragma unroll
  for (int k = 0; k < 6; ++k) sv[k] = __builtin_nontemporal_load(sp + k);
  v4f cv[3];
#pragma unroll
  for (int k = 0; k < 3; ++k) cv[k] = __builtin_nontemporal_load(cp + k);
  float st[24], ct[12];
#pragma unroll
  for (int k = 0; k < 6; ++k)
#pragma unroll
    for (int j = 0; j < 4; ++j) st[k * 4 + j] = sv[k][j];
#pragma unroll
  for (int k = 0; k < 3; ++k)
#pragma unroll
    for (int j = 0; j < 4; ++j) ct[k * 4 + j] = cv[k][j];
  float ot[24];
#pragma unroll
  for (int r = 0; r < 4; ++r)
    mecanum_row(st[6 * r + 2], st[6 * r + 3], st[6 * r + 4], st[6 * r + 5],
                ct[3 * r + 0], ct[3 * r + 1], ct[3 * r + 2], &ot[6 * r]);
  v4f* op = (v4f*)out + (size_t)i * 6;
#pragma unroll
  for (int k = 0; k < 6; ++k) {
    v4f v;
#pragma unroll
    for (int j = 0; j < 4; ++j) v[j] = ot[k * 4 + j];
    __builtin_nontemporal_store(v, op + k);
  }
}

// ---------------- scalar row tail ----------------
__global__ void mecanum_tail(const float* __restrict__ state, const float* __restrict__ ctrl,
                             float* __restrict__ out, int start, int B) {
  int r = start + blockIdx.x * blockDim.x + threadIdx.x;
  if (r >= B) return;
  const float* s = state + (size_t)r * 6;
  const float* u = ctrl  + (size_t)r * 3;
  float o[6];
  mecanum_row(s[2], s[3], s[4], s[5], u[0], u[1], u[2], o);
  float* op = out + (size_t)r * 6;
#pragma unroll
  for (int k = 0; k < 6; ++k) op[k] = o[k];
}

extern "C" void kernel_launch(void* const* d_in, const int* in_sizes, int n_in,
                              void* d_out, int out_size, void* d_ws, size_t ws_size,
                              hipStream_t stream) {
  // inputs: t(1), state(B*6), control_duty(B*3), l2w_rot(12), c2m(12), wt2la(12)
  const float* state = (const float*)d_in[1];
  const float* ctrl  = (const float*)d_in[2];
  float* out = (float*)d_out;

  long long B = (long long)in_sizes[1] / 6;
  long long nquad = B / 4;
  long long nfullwave = nquad / 32;              // 32-quad tiles (B=4M -> 31250 exact)

  if (nfullwave > 0) {
    int blocks = (int)((nfullwave + WAVES_PER_BLOCK - 1) / WAVES_PER_BLOCK);
    mecanum_async<<<blocks, WAVES_PER_BLOCK * 32, 0, stream>>>(state, ctrl, out,
                                                               (int)nfullwave);
  }
  long long startQuad = nfullwave * 32;
  long long remQuads = nquad - startQuad;
  if (remQuads > 0) {
    int blocks = (int)((remQuads + 255) / 256);
    mecanum_quad<<<blocks, 256, 0, stream>>>(state, ctrl, out, (int)startQuad, (int)nquad);
  }
  long long remRows = B - nquad * 4;
  if (remRows > 0) {
    mecanum_tail<<<1, 256, 0, stream>>>(state, ctrl, out, (int)(nquad * 4), (int)B);
  }
}